// ProposalTarget_75514114998601
// MI455X (gfx1250) — compile-verified
//
#include <hip/hip_runtime.h>
#include <math.h>
#include <stdint.h>

#define N_PROP      262144
#define N_GT        128
#define M_ALL       (N_PROP + N_GT)
#define NUM_CLASSES 21
#define ROIS        256
#define FG_ROIS     128

// ---------------------------------------------------------------------------
// Kernel 1: IoU max/argmax per ROI, fg/bg sampling keys.
// GT boxes (2KB) staged into LDS via CDNA5 async-to-LDS load; division-free
// fraction-compare argmax: in/un > bn/bd  <=>  in*bd > bn*un  (un > 0).
// fg test: bn/bd >= 0.5  <=>  2*bn >= bd.
// ---------------------------------------------------------------------------
__global__ __launch_bounds__(256) void iou_argmax_kernel(
    const float* __restrict__ proposals,   // [N_PROP,4]
    const float* __restrict__ gtb,         // [N_GT,4]
    const float* __restrict__ noise,       // [M_ALL]
    float* __restrict__ fgkey,             // [M_ALL]
    float* __restrict__ bgkey,             // [M_ALL]
    int*   __restrict__ assign)            // [M_ALL]
{
    __shared__ float4 sGT[N_GT];
    __shared__ float  sGA[N_GT];
    const int t = threadIdx.x;

    if (t < N_GT) {
        // wave-relative LDS byte offset = low 32 bits of the generic pointer
        unsigned lds_off = (unsigned)(uintptr_t)(&sGT[t]);
        unsigned long long ga = (unsigned long long)(uintptr_t)(gtb + 4 * t);
        asm volatile("global_load_async_to_lds_b128 %0, %1, off"
                     :: "v"(lds_off), "v"(ga) : "memory");
    }
    asm volatile("s_wait_asynccnt 0x0" ::: "memory");
    __syncthreads();

    if (t < N_GT) {
        float4 g = sGT[t];
        sGA[t] = (g.z - g.x) * (g.w - g.y);
    }
    __syncthreads();

    const int m = blockIdx.x * 256 + t;
    if (m >= M_ALL) return;

    float4 b;
    if (m < N_PROP) b = *reinterpret_cast<const float4*>(proposals + 4 * m);
    else            b = sGT[m - N_PROP];

    const float area_b = (b.z - b.x) * (b.w - b.y);

    // best IoU tracked as fraction bn/bd (bd > 0 after first iter)
    float bn = -1.0f, bd = 1.0f;
    int   besti = 0;
#pragma unroll 4
    for (int g = 0; g < N_GT; ++g) {
        float4 gb = sGT[g];
        float lx = fmaxf(b.x, gb.x);
        float ly = fmaxf(b.y, gb.y);
        float rx = fminf(b.z, gb.z);
        float ry = fminf(b.w, gb.w);
        float w  = fmaxf(rx - lx, 0.0f);
        float h  = fmaxf(ry - ly, 0.0f);
        float in = w * h;
        float un = (area_b - in) + sGA[g];      // > 0 (gt areas >= 64)
        bool better = (in * bd) > (bn * un);    // strict >: first max wins
        bn    = better ? in : bn;
        bd    = better ? un : bd;
        besti = better ? g  : besti;
    }

    const float nz = noise[m];
    const bool  fg = (2.0f * bn >= bd);          // max_iou >= 0.5
    fgkey[m]  = nz + (fg ? 2.0f : 0.0f);
    bgkey[m]  = nz + (fg ? 0.0f : 2.0f);
    assign[m] = besti;
}

// ---------------------------------------------------------------------------
// Kernel 2: exact top-128 per list via 4-level byte-radix select (256-bin LDS
// histogram per level => 4 scans instead of 32), then collect + bitonic sort
// of <=256 candidates. blockIdx.x==0 -> fg list, ==1 -> bg list.
// Keys (2x1MB) are L2-resident.
// ---------------------------------------------------------------------------
#define SEL_TPB 1024
#define SEL_CAP 256

__global__ __launch_bounds__(SEL_TPB) void select_topk_kernel(
    const float* __restrict__ fgkey,
    const float* __restrict__ bgkey,
    int* __restrict__ keep)                 // [256]: fg idx then bg idx
{
    const float* key = (blockIdx.x == 0) ? fgkey : bgkey;
    int* out = keep + blockIdx.x * FG_ROIS;

    __shared__ unsigned s_hist[256];
    __shared__ unsigned s_prefix;
    __shared__ unsigned s_krem;
    __shared__ unsigned s_ncol;
    __shared__ unsigned long long s_items[SEL_CAP];

    const int t = threadIdx.x;

    if (t == 0) { s_prefix = 0u; s_krem = FG_ROIS; }

    // --- radix select: find exact bit pattern of the 128th-largest key ---
    for (int shift = 24; shift >= 0; shift -= 8) {
        if (t < 256) s_hist[t] = 0u;
        __syncthreads();
        const unsigned prefix = s_prefix;
        const unsigned hmask  = (shift == 24) ? 0u : (0xFFFFFFFFu << (shift + 8));
        for (int i = t; i < M_ALL; i += SEL_TPB) {
            unsigned u = __float_as_uint(key[i]);
            if ((u & hmask) == (prefix & hmask))
                atomicAdd(&s_hist[(u >> shift) & 0xFFu], 1u);
        }
        __syncthreads();
        if (t == 0) {
            unsigned krem = s_krem, acc = 0u;
            int b = 255;
            for (; b > 0; --b) {
                if (acc + s_hist[b] >= krem) break;
                acc += s_hist[b];
            }
            s_prefix = prefix | ((unsigned)b << shift);
            s_krem   = krem - acc;
        }
        __syncthreads();
    }
    const unsigned thresh = s_prefix;   // value bits of 128th-largest key

    // --- collect candidates >= threshold (128 + rare tie slack) ---
    if (t == 0) s_ncol = 0u;
    __syncthreads();
    for (int i = t; i < M_ALL; i += SEL_TPB) {
        unsigned u = __float_as_uint(key[i]);
        if (u >= thresh) {
            unsigned p = atomicAdd(&s_ncol, 1u);
            if (p < (unsigned)SEL_CAP) {
                // pack: key desc major, index asc minor (top_k tie behavior)
                s_items[p] = ((unsigned long long)u << 32)
                           | (unsigned long long)(0x7FFFFFFFu - (unsigned)i);
            }
        }
    }
    __syncthreads();
    unsigned n = s_ncol; if (n > SEL_CAP) n = SEL_CAP;
    if (t < SEL_CAP && (unsigned)t >= n) s_items[t] = 0ull;  // pad low
    __syncthreads();

    // --- bitonic sort 256 items, descending ---
    for (unsigned k = 2; k <= SEL_CAP; k <<= 1) {
        for (unsigned j = k >> 1; j > 0; j >>= 1) {
            if (t < SEL_CAP) {
                unsigned ixj = (unsigned)t ^ j;
                if (ixj > (unsigned)t) {
                    unsigned long long a = s_items[t];
                    unsigned long long c = s_items[ixj];
                    bool descBlock = ((t & k) == 0);
                    bool doSwap = descBlock ? (a < c) : (a > c);
                    if (doSwap) { s_items[t] = c; s_items[ixj] = a; }
                }
            }
            __syncthreads();
        }
    }

    if (t < FG_ROIS)
        out[t] = (int)(0x7FFFFFFFu - (unsigned)(s_items[t] & 0xFFFFFFFFull));
}

// ---------------------------------------------------------------------------
// Kernel 3: gather 256 kept ROIs, bbox regression targets, class-slot scatter.
// Output layout (float): rois[256*4] | labels[256] | bbox_targets[256*84]
// bbox_targets written as 21 x b128 stores per row.
// ---------------------------------------------------------------------------
__global__ __launch_bounds__(ROIS) void finalize_kernel(
    const float* __restrict__ proposals,
    const float* __restrict__ gtb,
    const int*   __restrict__ labels,
    const float* __restrict__ fgkey,
    const int*   __restrict__ assign,
    const int*   __restrict__ keep,
    float* __restrict__ out)
{
    const int r = threadIdx.x;          // 0..255
    const int i = keep[r];

    float4 roi;
    if (i < N_PROP) roi = *reinterpret_cast<const float4*>(proposals + 4 * i);
    else            roi = *reinterpret_cast<const float4*>(gtb + 4 * (i - N_PROP));

    const int  a   = assign[i];
    const bool fg  = (fgkey[i] >= 2.0f);
    const int  lab = fg ? labels[a] : 0;

    const float4 g = *reinterpret_cast<const float4*>(gtb + 4 * a);

    // faster-rcnn bbox transform
    float ew  = roi.z - roi.x + 1.0f;
    float eh  = roi.w - roi.y + 1.0f;
    float ecx = roi.x + 0.5f * ew;
    float ecy = roi.y + 0.5f * eh;
    float gw  = g.z - g.x + 1.0f;
    float gh  = g.w - g.y + 1.0f;
    float gcx = g.x + 0.5f * gw;
    float gcy = g.y + 0.5f * gh;
    float dx  = (gcx - ecx) / ew;
    float dy  = (gcy - ecy) / eh;
    float dw  = logf(gw / ew);
    float dh  = logf(gh / eh);

    // rois
    out[r * 4 + 0] = roi.x;
    out[r * 4 + 1] = roi.y;
    out[r * 4 + 2] = roi.z;
    out[r * 4 + 3] = roi.w;
    // labels (bg -> 0)
    out[ROIS * 4 + r] = (float)lab;
    // bbox_targets: one float4 per class slot (aligned: 1280*4 + r*336 bytes)
    float4* bt4 = reinterpret_cast<float4*>(out + ROIS * 4 + ROIS
                                            + r * (4 * NUM_CLASSES));
    const float4 tgt  = make_float4(dx, dy, dw, dh);
    const float4 zero = make_float4(0.0f, 0.0f, 0.0f, 0.0f);
#pragma unroll
    for (int c = 0; c < NUM_CLASSES; ++c)
        bt4[c] = (fg && c == lab) ? tgt : zero;
}

// ---------------------------------------------------------------------------
extern "C" void kernel_launch(void* const* d_in, const int* in_sizes, int n_in,
                              void* d_out, int out_size, void* d_ws, size_t ws_size,
                              hipStream_t stream) {
    (void)in_sizes; (void)n_in; (void)out_size; (void)ws_size;
    const float* proposals = (const float*)d_in[0];   // [1,N_PROP,4] f32
    const float* gtb       = (const float*)d_in[1];   // [1,N_GT,4]  f32
    const int*   labels    = (const int*)  d_in[2];   // [1,N_GT]    i32
    const float* noise     = (const float*)d_in[3];   // [M_ALL]     f32

    float* fgkey  = (float*)d_ws;                 // M_ALL
    float* bgkey  = fgkey + M_ALL;                // M_ALL
    int*   assign = (int*)(bgkey + M_ALL);        // M_ALL
    int*   keep   = assign + M_ALL;               // 256

    float* out = (float*)d_out;

    iou_argmax_kernel<<<(M_ALL + 255) / 256, 256, 0, stream>>>(
        proposals, gtb, noise, fgkey, bgkey, assign);
    select_topk_kernel<<<2, SEL_TPB, 0, stream>>>(fgkey, bgkey, keep);
    finalize_kernel<<<1, ROIS, 0, stream>>>(
        proposals, gtb, labels, fgkey, assign, keep, out);
}